// VectorQuantizer_103079215403
// MI455X (gfx1250) — compile-verified
//
#include <hip/hip_runtime.h>
#include <hip/hip_bf16.h>

typedef __attribute__((ext_vector_type(2))) float v2f;
typedef __attribute__((ext_vector_type(4))) float v4f;
typedef __attribute__((ext_vector_type(8))) float v8f;

#define NUM_EMB 256
#define EMB_DIM 128
#define N_TOKENS 524288
#define CB_STRIDE 132            // padded dword stride (132 mod 64 == 4 -> conflict-free b64 pattern)
#define TOK_PER_WAVE 16
#define WAVES_PER_BLOCK 8
#define TOK_PER_BLOCK (TOK_PER_WAVE * WAVES_PER_BLOCK)   // 128

__global__ __launch_bounds__(256) void vq_wmma_kernel(
    const float* __restrict__ X, const float* __restrict__ W,
    float* __restrict__ out_loss, float* __restrict__ out_q,
    float* __restrict__ out_e, float* __restrict__ out_i)
{
    __shared__ float cb[NUM_EMB * CB_STRIDE];   // padded codebook
    __shared__ float w2s[NUM_EMB];              // ||w_k||^2
    __shared__ int   idx_s[TOK_PER_BLOCK];      // per-block argmin indices
    __shared__ float lsum[WAVES_PER_BLOCK];

    const int tid  = threadIdx.x;
    const int wave = tid >> 5;
    const int lane = tid & 31;
    const int lrow = lane & 15;   // row-in-tile / N slot
    const int half = lane >> 4;   // which K pair within a group of 4

    // ---- stage codebook into LDS (coalesced float4) ----
#pragma unroll
    for (int i = 0; i < 32; ++i) {
        int e4  = tid + i * 256;          // 8192 float4s total
        int row = e4 >> 5, c4 = e4 & 31;
        v4f v = *(const v4f*)&W[row * EMB_DIM + c4 * 4];
        *(v4f*)&cb[row * CB_STRIDE + c4 * 4] = v;
    }
    __syncthreads();
    {   // per-row squared norm (thread t -> row t)
        float s = 0.f;
        const float* r = &cb[tid * CB_STRIDE];
#pragma unroll 4
        for (int d = 0; d < EMB_DIM; ++d) { float x = r[d]; s += x * x; }
        w2s[tid] = s;
    }
    __syncthreads();

    const int token_base = blockIdx.x * TOK_PER_BLOCK + wave * TOK_PER_WAVE;

    // ---- load A fragments: 16 tokens x 128 dims, f32 16x16x4 A layout ----
    v2f a[32];
    const float* xrow = &X[(size_t)(token_base + lrow) * EMB_DIM + 2 * half];
#pragma unroll
    for (int k = 0; k < 32; ++k)
        a[k] = *(const v2f*)&xrow[k * 4];

    float minv[8]; int mini[8];
#pragma unroll
    for (int r = 0; r < 8; ++r) { minv[r] = 3.402823466e38f; mini[r] = 0; }

    // ---- scores for all 256 codes, 16 codes per column tile ----
    for (int ct = 0; ct < 16; ++ct) {
        const float* brow = &cb[(ct * 16 + lrow) * CB_STRIDE + 2 * half];
        v8f c = {0.f, 0.f, 0.f, 0.f, 0.f, 0.f, 0.f, 0.f};
#pragma unroll
        for (int k = 0; k < 32; ++k) {
            v2f b = *(const v2f*)&brow[k * 4];
            c = __builtin_amdgcn_wmma_f32_16x16x4_f32(false, a[k], false, b,
                                                      (short)0, c, false, false);
        }
        float w2v = w2s[ct * 16 + lrow];
        int   code = ct * 16 + lrow;
#pragma unroll
        for (int r = 0; r < 8; ++r) {
            float d = w2v - 2.f * c[r];           // ||x||^2 constant -> dropped
            if (d < minv[r]) { minv[r] = d; mini[r] = code; }
        }
    }

    // ---- argmin across the 16 lanes of each half (butterfly, low-index ties) ----
#pragma unroll
    for (int r = 0; r < 8; ++r) {
        float v = minv[r]; int i = mini[r];
#pragma unroll
        for (int off = 8; off >= 1; off >>= 1) {
            float ov = __shfl_xor(v, off, 32);
            int   oi = __shfl_xor(i, off, 32);
            if (ov < v || (ov == v && oi < i)) { v = ov; i = oi; }
        }
        if (lane == 0)       idx_s[wave * 16 + r]     = i;   // tokens M=0..7
        else if (lane == 16) idx_s[wave * 16 + 8 + r] = i;   // tokens M=8..15
    }
    __syncthreads();

    // ---- loss partial from register-resident A + LDS codebook ----
    float acc = 0.f;
    {
        int myidx = idx_s[wave * 16 + lrow];
        const float* qrow = &cb[myidx * CB_STRIDE + 2 * half];
#pragma unroll
        for (int k = 0; k < 32; ++k) {
            v2f q = *(const v2f*)&qrow[k * 4];
            float dx = q.x - a[k].x, dy = q.y - a[k].y;
            acc += dx * dx + dy * dy;
        }
    }

    // ---- write quantized rows + one-hot encodings ----
    for (int t = 0; t < 16; ++t) {
        int row = idx_s[wave * 16 + t];
        size_t tok = (size_t)(token_base + t);
        v4f qv = *(const v4f*)&cb[row * CB_STRIDE + lane * 4];
        *(v4f*)&out_q[tok * EMB_DIM + lane * 4] = qv;
        int cb0 = lane * 8;
        v4f e0, e1;
#pragma unroll
        for (int e = 0; e < 4; ++e) {
            e0[e] = (cb0 + e     == row) ? 1.f : 0.f;
            e1[e] = (cb0 + 4 + e == row) ? 1.f : 0.f;
        }
        *(v4f*)&out_e[tok * NUM_EMB + cb0]     = e0;
        *(v4f*)&out_e[tok * NUM_EMB + cb0 + 4] = e1;
    }
    if (lane < 16)
        out_i[token_base + lane] = (float)idx_s[wave * 16 + lane];

    // ---- loss reduction: wave -> block -> global atomic ----
#pragma unroll
    for (int off = 16; off >= 1; off >>= 1)
        acc += __shfl_xor(acc, off, 32);
    if (lane == 0) lsum[wave] = acc;
    __syncthreads();
    if (tid == 0) {
        float s = 0.f;
#pragma unroll
        for (int w = 0; w < WAVES_PER_BLOCK; ++w) s += lsum[w];
        // loss = 0.1*(q + 0.5*e) with q==e numerically => 0.15 * SSE / (N*D)
        atomicAdd(out_loss, s * (0.15f / 67108864.f));
    }
}

extern "C" void kernel_launch(void* const* d_in, const int* in_sizes, int n_in,
                              void* d_out, int out_size, void* d_ws, size_t ws_size,
                              hipStream_t stream) {
    (void)in_sizes; (void)n_in; (void)out_size; (void)d_ws; (void)ws_size;
    const float* X = (const float*)d_in[0];   // [524288, 128]
    const float* W = (const float*)d_in[1];   // [256, 128]
    float* out      = (float*)d_out;
    float* out_loss = out;                                    // [1]
    float* out_q    = out_loss + 1;                           // [N,128]
    float* out_e    = out_q + (size_t)N_TOKENS * EMB_DIM;     // [N,256]
    float* out_i    = out_e + (size_t)N_TOKENS * NUM_EMB;     // [N,1]

    hipMemsetAsync(out_loss, 0, sizeof(float), stream);
    dim3 grid(N_TOKENS / TOK_PER_BLOCK);
    vq_wmma_kernel<<<grid, 256, 0, stream>>>(X, W, out_loss, out_q, out_e, out_i);
}